// GNN_cont_65816078844127
// MI455X (gfx1250) — compile-verified
//
#include <hip/hip_runtime.h>

#define N_NODES 100000
#define N_EDGES 3200000
#define HID 128
#define PST 144   // pair-row stride in float2 units: 2*PST mod 64 == 32 dword-banks

typedef __attribute__((ext_vector_type(2))) float v2f;
typedef __attribute__((ext_vector_type(8))) float v8f;
typedef __attribute__((ext_vector_type(4))) int   v4i;

// Pointer-to-v4i in AMDGPU global (AS1) / LDS (AS3) address spaces.
typedef __attribute__((address_space(1))) v4i* gv4i_p;
typedef __attribute__((address_space(3))) v4i* lv4i_p;

// ---------------------------------------------------------------------------
// Indegree (step-invariant): indeg[d] += 1 per edge
// ---------------------------------------------------------------------------
__global__ void gnn_indeg_kernel(const int* __restrict__ dst,
                                 float* __restrict__ indeg, int nE) {
    int e = blockIdx.x * blockDim.x + threadIdx.x;
    if (e < nE) atomicAdd(&indeg[dst[e]], 1.0f);
}

// ---------------------------------------------------------------------------
// Edge scatter: agg[dst] += h[src], 128 floats per edge.
// One wave (32 lanes) per edge; each lane moves a float4 (coalesced 512B).
// ---------------------------------------------------------------------------
__global__ void gnn_scatter_kernel(const int* __restrict__ src,
                                   const int* __restrict__ dst,
                                   const float* __restrict__ h,
                                   float* __restrict__ agg, int nE) {
    unsigned gid = blockIdx.x * 256u + threadIdx.x;
    unsigned e   = gid >> 5;
    if (e >= (unsigned)nE) return;
    int l = gid & 31;
    int s = src[e];
    int d = dst[e];
    const float4 v = *reinterpret_cast<const float4*>(h + (size_t)s * HID + l * 4);
    float* p = agg + (size_t)d * HID + l * 4;
    atomicAdd(p + 0, v.x);
    atomicAdd(p + 1, v.y);
    atomicAdd(p + 2, v.z);
    atomicAdd(p + 3, v.w);
}

// ---------------------------------------------------------------------------
// K-loop helper: 32 x V_WMMA_F32_16X16X4_F32 over K=128.
// Uniform template split on HAS_A1 keeps EXEC untouched inside the loop.
// B is pair-interleaved in LDS: Bp[k/2][col] = {B[k][col], B[k+1][col]},
// so each B-fragment is ONE aligned ds_load_b64 into a consecutive VGPR
// pair (no operand-repack movs), conflict-free across both lane halves.
// ---------------------------------------------------------------------------
template <bool HAS_A1>
__device__ __forceinline__ void gnn_k_loop(const float* __restrict__ a0p,
                                           const float* __restrict__ a1p,
                                           const float2* __restrict__ Bp2,
                                           int cl, int koff, v8f (&acc)[8]) {
    for (int k0 = 0; k0 < HID; k0 += 4) {
        v2f a;
        const float2 x0 = *reinterpret_cast<const float2*>(a0p + k0 + koff);
        if (HAS_A1) {
            const float2 x1 = *reinterpret_cast<const float2*>(a1p + k0 + koff);
            a = (v2f){x0.x + x1.x, x0.y + x1.y};   // A = h + agg (layer 1)
        } else {
            a = (v2f){x0.x, x0.y};
        }
        const float2* bp = Bp2 + (size_t)((k0 + koff) >> 1) * PST + cl;
#pragma unroll
        for (int nt = 0; nt < 8; ++nt) {
            v2f b = *reinterpret_cast<const v2f*>(bp + nt * 16);
            acc[nt] = __builtin_amdgcn_wmma_f32_16x16x4_f32(
                false, a, false, b, (short)0, acc[nt], false, false);
        }
    }
}

// ---------------------------------------------------------------------------
// Generic WMMA fp32 GEMM:  out = f( (A0 [+A1]) @ W + bias[col] (+ time term) )
//   mode 0: out = acc                               (embedding, bias=emb_b)
//   mode 1: out = relu(acc), C-init += t*(1+indeg[m])*w1row0[col]
//   mode 2: out = hprev + dt*acc                    (Euler update)
// W [128,128] row-major, pair-interleaved into LDS. bias / w1row0 vectors
// staged via GLOBAL_LOAD_ASYNC_TO_LDS_B128 (ASYNCcnt path) when available.
// Block = 8 waves; each wave owns 16 rows x 128 cols (8 16x16 accumulators).
// ---------------------------------------------------------------------------
__global__ __launch_bounds__(256) void gnn_wmma_gemm(
    const float* __restrict__ A0, const float* __restrict__ A1,
    const float* __restrict__ W,  const float* __restrict__ bias,
    const float* __restrict__ w1row0, const float* __restrict__ indeg,
    const float* __restrict__ hprev, float* __restrict__ out,
    float t, float dt, int n, int mode)
{
    __shared__ float2 Bp[64 * PST];   // 73.7 KB pair-interleaved weights
    __shared__ float  BiasS[HID];
    __shared__ float  W0S[HID];

    const int tid = threadIdx.x;

    // --- Stage bias (and W1 row 0 for mode 1) via CDNA5 async global->LDS ---
#if __has_builtin(__builtin_amdgcn_global_load_async_to_lds_b128)
    {
        float* bnc = const_cast<float*>(bias);
        if (tid < 32)
            __builtin_amdgcn_global_load_async_to_lds_b128(
                (gv4i_p)(bnc + tid * 4), (lv4i_p)&BiasS[tid * 4], 0, 0);
        if (mode == 1 && tid >= 32 && tid < 64) {
            float* wnc = const_cast<float*>(w1row0);
            __builtin_amdgcn_global_load_async_to_lds_b128(
                (gv4i_p)(wnc + (tid - 32) * 4), (lv4i_p)&W0S[(tid - 32) * 4], 0, 0);
        }
    }
#else
    if (tid < HID) {
        BiasS[tid] = bias[tid];
        if (mode == 1) W0S[tid] = w1row0[tid];
    }
#endif

    // --- Pair-interleave W into LDS: Bp[k2][c] = {W[2k2][c], W[2k2+1][c]} ---
    for (int q = tid; q < 64 * 64; q += 256) {
        int k2 = q >> 6;            // K pair index 0..63
        int c2 = (q & 63) << 1;     // even column
        const float2 r0 = *reinterpret_cast<const float2*>(W + (size_t)(2 * k2)     * HID + c2);
        const float2 r1 = *reinterpret_cast<const float2*>(W + (size_t)(2 * k2 + 1) * HID + c2);
        *reinterpret_cast<float4*>(&Bp[(size_t)k2 * PST + c2]) =
            make_float4(r0.x, r1.x, r0.y, r1.y);
    }

#if __has_builtin(__builtin_amdgcn_global_load_async_to_lds_b128)
#if __has_builtin(__builtin_amdgcn_s_wait_asynccnt)
    __builtin_amdgcn_s_wait_asynccnt(0);
#else
    asm volatile("s_wait_asynccnt 0x0" ::: "memory");
#endif
#endif
    __syncthreads();

    const int wave  = tid >> 5;
    const int lane  = tid & 31;
    const int cl    = lane & 15;         // column-within-tile / row-within-tile
    const int hi    = lane >> 4;         // lane half
    const int koff  = hi * 2;            // K sub-offset for A/B frags
    const int tileM = blockIdx.x * 128 + wave * 16;

    int rowA = tileM + cl;               // A-frag row for this lane
    if (rowA >= n) rowA = n - 1;         // clamp loads (stores predicated later)
    const float* a0p = A0 + (size_t)rowA * HID;
    const float* a1p = A1 + (size_t)rowA * HID;   // valid only when A1 != null

    // --- C initialization (bias, optionally + t*(1+indeg)*W1row0) ---
    float tc[8];
    if (mode == 1) {
#pragma unroll
        for (int r = 0; r < 8; ++r) {
            int m = tileM + r + (hi << 3);
            int mc = m < n ? m : n - 1;
            tc[r] = t * (1.0f + indeg[mc]);
        }
    }
    v8f acc[8];
#pragma unroll
    for (int nt = 0; nt < 8; ++nt) {
        int col = nt * 16 + cl;
        float bv = BiasS[col];
        if (mode == 1) {
            float w0 = W0S[col];
#pragma unroll
            for (int r = 0; r < 8; ++r) acc[nt][r] = bv + tc[r] * w0;
        } else {
#pragma unroll
            for (int r = 0; r < 8; ++r) acc[nt][r] = bv;
        }
    }

    // --- K loop (uniform branch on kernarg pointer; no EXEC churn inside) ---
    if (A1 != nullptr) gnn_k_loop<true >(a0p, a1p, Bp, cl, koff, acc);
    else               gnn_k_loop<false>(a0p, a1p, Bp, cl, koff, acc);

    // --- Epilogue + store (predicated per row) ---
#pragma unroll
    for (int nt = 0; nt < 8; ++nt) {
#pragma unroll
        for (int r = 0; r < 8; ++r) {
            int m = tileM + r + (hi << 3);
            if (m >= n) continue;
            int col = nt * 16 + cl;
            float v = acc[nt][r];
            if (mode == 1) v = v > 0.0f ? v : 0.0f;                      // ReLU
            if (mode == 2) v = hprev[(size_t)m * HID + col] + dt * v;    // Euler
            out[(size_t)m * HID + col] = v;
        }
    }
}

// ---------------------------------------------------------------------------
extern "C" void kernel_launch(void* const* d_in, const int* in_sizes, int n_in,
                              void* d_out, int out_size, void* d_ws, size_t ws_size,
                              hipStream_t stream) {
    const float* x    = (const float*)d_in[0];
    const int*   eidx = (const int*)  d_in[1];   // [2, N_EDGES]
    const float* embW = (const float*)d_in[2];   // [128,128]
    const float* embb = (const float*)d_in[3];   // [128]
    const float* W1   = (const float*)d_in[4];   // [129,128] row-major
    const float* b1   = (const float*)d_in[5];   // [128]
    const float* W2   = (const float*)d_in[6];   // [128,128]
    const float* b2   = (const float*)d_in[7];   // [128]
    float* out = (float*)d_out;                  // [5, N, 128]; slice i = h_i

    const int* src = eidx;
    const int* dst = eidx + N_EDGES;

    float* agg   = (float*)d_ws;                        // [N,128]
    float* u     = agg + (size_t)N_NODES * HID;         // [N,128]
    float* indeg = u   + (size_t)N_NODES * HID;         // [N]

    // Step-invariant indegree
    (void)hipMemsetAsync(indeg, 0, N_NODES * sizeof(float), stream);
    gnn_indeg_kernel<<<(N_EDGES + 255) / 256, 256, 0, stream>>>(dst, indeg, N_EDGES);

    const dim3 gemm_grid((N_NODES + 127) / 128);

    // h0 = x @ emb_W + emb_b  -> out slice 0
    gnn_wmma_gemm<<<gemm_grid, 256, 0, stream>>>(
        x, nullptr, embW, embb, nullptr, nullptr, nullptr,
        out, 0.0f, 0.0f, N_NODES, 0);

    const unsigned scatter_blocks = (unsigned)(((size_t)N_EDGES * 32 + 255) / 256);
    for (int i = 0; i < 4; ++i) {
        const float t  = 0.25f * (float)i;   // ts = linspace(0,1,5)
        const float dt = 0.25f;
        float* h  = out + (size_t)i       * N_NODES * HID;
        float* hn = out + (size_t)(i + 1) * N_NODES * HID;

        // agg = segment_sum(h[src], dst)
        (void)hipMemsetAsync(agg, 0, (size_t)N_NODES * HID * sizeof(float), stream);
        gnn_scatter_kernel<<<scatter_blocks, 256, 0, stream>>>(src, dst, h, agg, N_EDGES);

        // u = relu((h+agg) @ W1[1:,:] + b1 + t*(1+indeg)*W1[0,:])
        gnn_wmma_gemm<<<gemm_grid, 256, 0, stream>>>(
            h, agg, W1 + HID, b1, W1 /*row 0*/, indeg, nullptr,
            u, t, 0.0f, N_NODES, 1);

        // h_next = h + dt * (u @ W2 + b2)
        gnn_wmma_gemm<<<gemm_grid, 256, 0, stream>>>(
            u, nullptr, W2, b2, nullptr, nullptr, h,
            hn, 0.0f, dt, N_NODES, 2);
    }
}